// BeamSearch_61375082660509
// MI455X (gfx1250) — compile-verified
//
#include <hip/hip_runtime.h>
#include <hip/hip_bf16.h>

#define PAD_ID 1
#define VOCAB  50257
#define NBEAM  8
#define BSZ    64
#define VK     16
#define TPB    256
#define NWAVE  (TPB / 32)
#define TILE   1024   /* floats per tile = TPB * 4 */
#define CHUNKS 2      /* blocks per (row,beam) for occupancy */
#define NLISTS (NBEAM * CHUNKS)

// ---- CDNA5 async global->LDS path (guarded; falls back to plain loads) ----
#if defined(__HIP_DEVICE_COMPILE__) && __has_builtin(__builtin_amdgcn_global_load_async_to_lds_b128)
#define HAVE_ASYNC_LDS 1
#else
#define HAVE_ASYNC_LDS 0
#endif

#if defined(__HIP_DEVICE_COMPILE__) && __has_builtin(__builtin_amdgcn_s_wait_asynccnt)
#define HAVE_WAIT_ASYNC 1
#else
#define HAVE_WAIT_ASYNC 0
#endif

typedef int v4i __attribute__((ext_vector_type(4)));
typedef __attribute__((address_space(1))) v4i as1_v4i;   // global-AS int4
typedef __attribute__((address_space(3))) v4i as3_v4i;   // LDS-AS int4

__device__ __forceinline__ void stage16(const float4* g, float4* l) {
#if HAVE_ASYNC_LDS
  // global_load_async_to_lds_b128: per-lane 16B DMA, ASYNCcnt-tracked.
  // cpol=1 -> TH_NT: single-use stream, don't retain in caches.
  __builtin_amdgcn_global_load_async_to_lds_b128(
      (as1_v4i*)g, (as3_v4i*)l, 0, 1);
#else
  *l = *g;
#endif
}

template <int N>
__device__ __forceinline__ void wait_async_le() {
#if HAVE_ASYNC_LDS
#if HAVE_WAIT_ASYNC
  __builtin_amdgcn_s_wait_asynccnt((unsigned short)N);
  asm volatile("" ::: "memory");
#else
  asm volatile("s_wait_asynccnt %0" :: "i"(N) : "memory");
#endif
#endif
}

__device__ __forceinline__ void lds_drain_fence() {
#if HAVE_ASYNC_LDS
  // ensure our ds_load of the buffer completed before DMA overwrites it
  asm volatile("s_wait_dscnt 0x0" ::: "memory");
#endif
}

// ---- monotone f32 <-> u32 key mapping ----
__device__ __forceinline__ unsigned int sortable_f32(float x) {
  unsigned int u = __float_as_uint(x);
  return u ^ ((unsigned int)((int)u >> 31) | 0x80000000u);
}
__device__ __forceinline__ float unsortable_f32(unsigned int s) {
  return __uint_as_float((s & 0x80000000u) ? (s ^ 0x80000000u) : ~s);
}

// wave32 butterfly max over u64
__device__ __forceinline__ unsigned long long wave_max_u64(unsigned long long v) {
#pragma unroll
  for (int m = 16; m >= 1; m >>= 1) {
    unsigned long long o = __shfl_xor(v, m, 32);
    v = (o > v) ? o : v;
  }
  return v;
}

// =====================================================================
// Kernel 1: per-(row,beam,chunk) top-16.
// grid = BSZ*NBEAM*CHUNKS, block = 256 (8 wave32s)
// =====================================================================
__global__ __launch_bounds__(TPB) void bs_partial_topk(
    const float* __restrict__ lprobs,
    const float* __restrict__ scores,
    const int*   __restrict__ stepp,
    unsigned long long* __restrict__ partial) {
  __shared__ float4 lbuf[2][TPB];
  __shared__ unsigned long long wbuf[NWAVE][VK];

  const int blk    = (int)blockIdx.x;
  const int p      = blk / CHUNKS;          // row*NBEAM + beam
  const int chunk  = blk - p * CHUNKS;
  const int beamId = p & (NBEAM - 1);
  const int tid    = (int)threadIdx.x;
  const int lane   = tid & 31;
  const int wid    = tid >> 5;
  const int step   = *stepp;

  if (step == 0 && beamId != 0) {           // step 0: only beam 0 contributes
    if (tid < VK) partial[(size_t)blk * VK + tid] = 0ull;
    return;
  }

  const float score = (step == 0) ? 0.0f : scores[p];
  const float* base = lprobs + (size_t)p * VOCAB;
  const unsigned int gbase = (unsigned int)beamId * VOCAB;

  // chunk token range [lo, hi)
  const int lo = (VOCAB * chunk) / CHUNKS;
  const int hi = (VOCAB * (chunk + 1)) / CHUNKS;

  unsigned long long tk[VK];
#pragma unroll
  for (int i = 0; i < VK; ++i) tk[i] = 0ull;
  unsigned int thresh = 0u;                 // hi32 of current min key

  auto consider = [&](int token, float v) {
    if (token == PAD_ID) return;            // reference masks PAD to -inf
    const unsigned int sv = sortable_f32(v + score);
    if (sv < thresh) return;                // fast 32-bit reject (common case)
    const unsigned long long key =
        ((unsigned long long)sv << 32) |
        (unsigned long long)(0xFFFFFFFFu - (gbase + (unsigned int)token));
    if (key > tk[VK - 1]) {                 // rare insert: one bubble pass
      tk[VK - 1] = key;
#pragma unroll
      for (int i = VK - 1; i > 0; --i) {
        if (tk[i] > tk[i - 1]) {
          const unsigned long long t2 = tk[i - 1];
          tk[i - 1] = tk[i];
          tk[i] = t2;
        }
      }
      thresh = (unsigned int)(tk[VK - 1] >> 32);
    }
  };

  // element index p*VOCAB+lo is only 4B aligned; peel head to 16B alignment
  const int h      = (4 - ((p + lo) & 3)) & 3;     // VOCAB % 4 == 1
  const int NT     = (hi - lo - h) / TILE;
  const int tstart = lo + h + NT * TILE;
  const float4* gsrc = (const float4*)(base + lo + h);

  // 2-deep async double-buffer: each lane DMAs and consumes its own 16B,
  // so wave-level s_wait_asynccnt is sufficient ordering (no barriers).
  if (NT > 0) stage16(&gsrc[tid], &lbuf[0][tid]);
  if (NT > 1) stage16(&gsrc[TPB + tid], &lbuf[1][tid]);

  for (int t = 0; t < NT; ++t) {
    if (t + 1 < NT) wait_async_le<1>(); else wait_async_le<0>();
    const float4 v = lbuf[t & 1][tid];
    const int tok0 = lo + h + t * TILE + (tid << 2);
    consider(tok0 + 0, v.x);
    consider(tok0 + 1, v.y);
    consider(tok0 + 2, v.z);
    consider(tok0 + 3, v.w);
    if (t + 2 < NT) {
      lds_drain_fence();
      stage16(&gsrc[(size_t)(t + 2) * TPB + tid], &lbuf[t & 1][tid]);
    }
  }

  // head + tail scalars
  for (int token = lo + tid; token < lo + h; token += TPB) consider(token, base[token]);
  for (int token = tstart + tid; token < hi; token += TPB) consider(token, base[token]);

  // ---- wave merge: 32 sorted lists -> wave top-16 (16 rounds of max) ----
  unsigned long long cand = tk[0];
  for (int r = 0; r < VK; ++r) {
    const unsigned long long m = wave_max_u64(cand);
    const unsigned long long ball = __ballot(cand == m);
    const int first = __ffsll(ball) - 1;
    if (lane == first) {                    // winner pops its head
#pragma unroll
      for (int i = 0; i < VK - 1; ++i) tk[i] = tk[i + 1];
      tk[VK - 1] = 0ull;
      cand = tk[0];
    }
    if (lane == 0) wbuf[wid][r] = m;
  }
  __syncthreads();

  // ---- block merge: 8 sorted wave lists -> block top-16 ----
  if (wid == 0) {
    int ptr = 0;
    unsigned long long c2 = (lane < NWAVE) ? wbuf[lane][0] : 0ull;
    for (int r = 0; r < VK; ++r) {
      const unsigned long long m = wave_max_u64(c2);
      const unsigned long long ball = __ballot(c2 == m);
      const int first = __ffsll(ball) - 1;
      if (lane == first) {
        ++ptr;
        c2 = (ptr < VK && lane < NWAVE) ? wbuf[lane][ptr] : 0ull;
      }
      if (lane == 0) partial[(size_t)blk * VK + r] = m;
    }
  }
}

// =====================================================================
// Kernel 2: merge 16 sorted lists per row -> final top-16.
// grid = BSZ, block = 32 (one wave)
// =====================================================================
__global__ __launch_bounds__(32) void bs_final_merge(
    const unsigned long long* __restrict__ partial,
    float* __restrict__ out) {
  const int row  = (int)blockIdx.x;
  const int lane = (int)threadIdx.x;
  const unsigned long long* lists = partial + (size_t)row * NLISTS * VK;

  int ptr = 0;
  unsigned long long cand = (lane < NLISTS) ? lists[(size_t)lane * VK] : 0ull;
  for (int r = 0; r < VK; ++r) {
    const unsigned long long m = wave_max_u64(cand);
    const unsigned long long ball = __ballot(cand == m);
    const int first = __ffsll(ball) - 1;
    if (lane == first) {
      ++ptr;
      cand = (ptr < VK && lane < NLISTS) ? lists[(size_t)lane * VK + ptr] : 0ull;
    }
    if (lane == 0) {
      const unsigned int hi   = (unsigned int)(m >> 32);
      const unsigned int gidx = 0xFFFFFFFFu - (unsigned int)(m & 0xFFFFFFFFu);
      const unsigned int beamIdx = gidx / VOCAB;
      const unsigned int token   = gidx - beamIdx * VOCAB;
      out[(size_t)row * VK + r]                    = unsortable_f32(hi);   // scores_buf
      out[(size_t)BSZ * VK + (size_t)row * VK + r] = (float)token;         // indices_buf
      out[2u * BSZ * VK + (size_t)row * VK + r]    = (float)beamIdx;       // beams_buf
    }
  }
}

extern "C" void kernel_launch(void* const* d_in, const int* in_sizes, int n_in,
                              void* d_out, int out_size, void* d_ws, size_t ws_size,
                              hipStream_t stream) {
  (void)in_sizes; (void)n_in; (void)out_size; (void)ws_size;
  const float* lprobs = (const float*)d_in[0];
  const float* scores = (const float*)d_in[1];
  const int*   step   = (const int*)d_in[2];
  unsigned long long* partial = (unsigned long long*)d_ws;  // 1024*16*8B = 128 KB
  float* out = (float*)d_out;

  bs_partial_topk<<<BSZ * NBEAM * CHUNKS, TPB, 0, stream>>>(lprobs, scores, step, partial);
  bs_final_merge<<<BSZ, 32, 0, stream>>>(partial, out);
}